// YOLOv2Loss_62569083568619
// MI455X (gfx1250) — compile-verified
//
#include <hip/hip_runtime.h>

// YOLOv2 loss for MI455X (gfx1250, wave32).
// Stage 1: one block per image; LDS-staged GT state; streaming pass over the
//          1445 (cell,anchor) slots; per-wave reduction via V_WMMA_F32_16X16X4_F32.
// Stage 2: deterministic final reduction of per-block sums (also via WMMA).

namespace {

constexpr int kS  = 17;
constexpr int kA  = 5;
constexpr int kC  = 20;
constexpr int kG  = 32;
constexpr int kHW = kS * kS;       // 289
constexpr int kNLoc = kHW * kA;    // 1445
constexpr float kIouThresh  = 0.6f;
constexpr float kObjScale   = 5.0f;
constexpr float kCoordNoobj = 0.01f;

typedef __attribute__((ext_vector_type(2))) float v2f;
typedef __attribute__((ext_vector_type(8))) float v8f;

// Full-wave (32 lane) fp32 sum using one WMMA: D = A(16x4) * ones(4x16).
// A holds each lane's partial exactly once (VGPR0 = partial, VGPR1 = 0), so
// every row-sum of D is a subset sum and Σ rows = Σ lane partials.
// Lanes 0..15 hold rows 0..7 in their 8 D VGPRs, lanes 16..31 hold rows 8..15;
// summing the 8 VGPRs then xor-16 shuffle-adding yields the wave total in all lanes.
__device__ inline float wave_sum_wmma(float partial) {
  v2f a;    a[0] = partial; a[1] = 0.0f;
  v2f ones; ones[0] = 1.0f; ones[1] = 1.0f;
  v8f acc = {};
  acc = __builtin_amdgcn_wmma_f32_16x16x4_f32(
      /*neg_a=*/false, a, /*neg_b=*/false, ones,
      /*c_mod=*/(short)0, acc, /*reuse_a=*/false, /*reuse_b=*/false);
  float s = acc[0] + acc[1] + acc[2] + acc[3] + acc[4] + acc[5] + acc[6] + acc[7];
  s += __shfl_xor(s, 16, 32);
  return s;
}

__global__ __launch_bounds__(256) void yolo_loss_stage1(
    const float* __restrict__ bbox,    // (B, HW, A, 4)
    const float* __restrict__ ioup,    // (B, HW, A, 1)
    const float* __restrict__ score,   // (B, HW, A, C)
    const float* __restrict__ tgt,     // (B*G, 6)
    const float* __restrict__ anchors, // (A, 2)
    float* __restrict__ block_sums)    // (B,)
{
  __shared__ float s_anc[2 * kA];                  // anchors / 17
  __shared__ float s_gx1[kG], s_gy1[kG], s_gx2[kG], s_gy2[kG];
  __shared__ float s_tar[kG][4];                   // (offx, offy, w, h)
  __shared__ int   s_cls[kG];
  __shared__ short s_amap[kNLoc];                  // slot -> gt index or -1
  __shared__ float s_wave[8];

  const int b   = blockIdx.x;
  const int tid = threadIdx.x;

  for (int i = tid; i < kNLoc; i += blockDim.x) s_amap[i] = -1;
  if (tid < 2 * kA) s_anc[tid] = anchors[tid] / 17.0f;
  __syncthreads();

  // ---- GT preprocessing: one lane per GT box (cells are distinct per image) ----
  if (tid < kG) {
    const float* t = tgt + (long)(b * kG + tid) * 6;
    const int   cls = (int)t[1];
    const float x1 = t[2], y1 = t[3], x2 = t[4], y2 = t[5];
    const float cx = 0.5f * (x1 + x2), cy = 0.5f * (y1 + y2);
    const float w  = x2 - x1,          h  = y2 - y1;
    const float sx = cx * (float)kS,   sy = cy * (float)kS;
    const int   ci = (int)floorf(sx),  cj = (int)floorf(sy);
    const float offx = sx - (float)ci, offy = sy - (float)cj;
    const int   cell = ci * kS + cj;

    // anchor assignment: argmax_a IoU(anchor_a, gt_wh), first index wins ties
    const float g_area = w * h;
    float best = -1.0f; int aidx = 0;
    for (int a = 0; a < kA; ++a) {
      const float ax = s_anc[2 * a], ay = s_anc[2 * a + 1];
      const float inter = fminf(ax, w) * fminf(ay, h);
      const float aiou  = inter / (ax * ay + g_area - inter);
      if (aiou > best) { best = aiou; aidx = a; }
    }

    s_gx1[tid] = x1; s_gy1[tid] = y1; s_gx2[tid] = x2; s_gy2[tid] = y2;
    s_tar[tid][0] = offx; s_tar[tid][1] = offy; s_tar[tid][2] = w; s_tar[tid][3] = h;
    s_cls[tid] = cls;
    s_amap[cell * kA + aidx] = (short)tid;
  }
  __syncthreads();

  // ---- Streaming pass over the 1445 (cell, anchor) slots ----
  float partial = 0.0f;
  const long base = (long)b * kNLoc;

  for (int loc = tid; loc < kNLoc; loc += blockDim.x) {
    const int hw = loc / kA, a  = loc - hw * kA;
    const int gi = hw / kS,  gj = hw - gi * kS;

    const float4 p  = *(const float4*)(bbox + (base + loc) * 4);
    const float  ip = ioup[base + loc];

    // decode to corner box (matches reference op order: anc = anchors/17, /S)
    const float x = (p.x + (float)gi) / (float)kS;
    const float y = (p.y + (float)gj) / (float)kS;
    const float w = p.z * s_anc[2 * a]     / (float)kS;
    const float h = p.w * s_anc[2 * a + 1] / (float)kS;
    const float px1 = x - 0.5f * w, py1 = y - 0.5f * h;
    const float px2 = x + 0.5f * w, py2 = y + 0.5f * h;
    const float pa  = (px2 - px1) * (py2 - py1);   // corner-derived, like reference

    const short ag = s_amap[loc];
    float best_iou = 0.0f;
    float iou_t    = 0.0f;
    for (int g = 0; g < kG; ++g) {
      const float lx = fmaxf(px1, s_gx1[g]), ly = fmaxf(py1, s_gy1[g]);
      const float rx = fminf(px2, s_gx2[g]), ry = fminf(py2, s_gy2[g]);
      const float iw = fmaxf(rx - lx, 0.0f), ih = fmaxf(ry - ly, 0.0f);
      const float inter = iw * ih;
      const float ga  = (s_gx2[g] - s_gx1[g]) * (s_gy2[g] - s_gy1[g]);
      const float iou = inter / (pa + ga - inter);
      best_iou = fmaxf(best_iou, iou);
      if (g == (int)ag) iou_t = iou;
    }

    // IoU loss term: (ip*mask - iou_t*mask)^2. Assigned override wins.
    const float imask = (ag >= 0) ? kObjScale * (1.0f - ip)
                                  : ((best_iou <= kIouThresh) ? -ip : 0.0f);
    const float itv = (ag >= 0) ? iou_t : 0.0f;
    const float di  = (ip - itv) * imask;
    partial += di * di;

    // Box loss term (raw preds vs target; default target (0.5,0.5,1,1), mask 0.01)
    float bm, t0, t1, t2, t3;
    if (ag >= 0) {
      bm = 1.0f;
      t0 = s_tar[ag][0]; t1 = s_tar[ag][1]; t2 = s_tar[ag][2]; t3 = s_tar[ag][3];
    } else {
      bm = kCoordNoobj; t0 = 0.5f; t1 = 0.5f; t2 = 1.0f; t3 = 1.0f;
    }
    const float d0 = (p.x - t0) * bm, d1 = (p.y - t1) * bm;
    const float d2 = (p.z - t2) * bm, d3 = (p.w - t3) * bm;
    partial += d0 * d0 + d1 * d1 + d2 * d2 + d3 * d3;

    // Class loss: mask is zero except at assigned slots -> only read those rows.
    if (ag >= 0) {
      const int gc = s_cls[ag];
      const float* sp = score + (base + loc) * kC;   // 80B offset, 16B aligned
      for (int c0 = 0; c0 < kC; c0 += 4) {
        const float4 sv = *(const float4*)(sp + c0);
        const float e0 = sv.x - ((c0 + 0) == gc ? 1.0f : 0.0f);
        const float e1 = sv.y - ((c0 + 1) == gc ? 1.0f : 0.0f);
        const float e2 = sv.z - ((c0 + 2) == gc ? 1.0f : 0.0f);
        const float e3 = sv.w - ((c0 + 3) == gc ? 1.0f : 0.0f);
        partial += e0 * e0 + e1 * e1 + e2 * e2 + e3 * e3;
      }
    }
  }

  // ---- Block reduction: WMMA per wave, then fixed-order wave combine ----
  const float s = wave_sum_wmma(partial);          // EXEC all-ones here
  if ((tid & 31) == 0) s_wave[tid >> 5] = s;
  __syncthreads();
  if (tid == 0) {
    float t = 0.0f;
    for (int i = 0; i < 8; ++i) t += s_wave[i];
    block_sums[b] = t * (1.0f / (float)kG);        // /G applied once per image
  }
}

__global__ __launch_bounds__(256) void yolo_loss_stage2(
    const float* __restrict__ block_sums, int n, float* __restrict__ out)
{
  __shared__ float s_wave[8];
  const int tid = threadIdx.x;
  float partial = 0.0f;
  for (int i = tid; i < n; i += 256) partial += block_sums[i];
  const float s = wave_sum_wmma(partial);
  if ((tid & 31) == 0) s_wave[tid >> 5] = s;
  __syncthreads();
  if (tid == 0) {
    float t = 0.0f;
    for (int i = 0; i < 8; ++i) t += s_wave[i];
    out[0] = t;
  }
}

} // namespace

extern "C" void kernel_launch(void* const* d_in, const int* in_sizes, int n_in,
                              void* d_out, int out_size, void* d_ws, size_t ws_size,
                              hipStream_t stream) {
  const float* bbox    = (const float*)d_in[0];
  const float* ioup    = (const float*)d_in[1];
  const float* score   = (const float*)d_in[2];
  const float* tgt     = (const float*)d_in[3];
  const float* anchors = (const float*)d_in[4];
  float* out  = (float*)d_out;
  float* bsum = (float*)d_ws;                       // nb floats of scratch

  const int nb = in_sizes[0] / (kNLoc * 4);         // batch size from bbox_pred

  yolo_loss_stage1<<<nb, 256, 0, stream>>>(bbox, ioup, score, tgt, anchors, bsum);
  yolo_loss_stage2<<<1, 256, 0, stream>>>(bsum, nb, out);
}